// UniMPBlock_188978561506
// MI455X (gfx1250) — compile-verified
//
#include <hip/hip_runtime.h>
#include <hip/hip_bf16.h>
#include <math.h>

typedef float v2f __attribute__((ext_vector_type(2)));
typedef float v8f __attribute__((ext_vector_type(8)));

#define D_DIM   128
#define HC_DIM  512
#define H_HEADS 4

// ---------- ordered-uint encoding for float atomic max ----------
__device__ __forceinline__ unsigned encOrd(float f) {
    int b = __float_as_int(f);
    return (b >= 0) ? (unsigned(b) | 0x80000000u) : unsigned(~b);
}
__device__ __forceinline__ float decOrd(unsigned u) {
    int b = (u & 0x80000000u) ? int(u & 0x7FFFFFFFu) : ~int(u);
    return __int_as_float(b);
}

// ---------------------------------------------------------------
// fp32 WMMA GEMM:  C[M,NC] = A[M,K] @ W[K,NC] + bias
// K, NC compile-time -> B addresses fold to base + immediate
// IOFFSET, inner 64-K block fully unrolled (16 WMMAs back to back,
// loads batched ahead). block = 256 (8 waves); blockIdx.x = 16-row
// M tile, blockIdx.y = 128-col N tile. A tile staged in padded LDS.
// MODE==1: fused epilogue  out = elu(C + resid), resid row-stride 128.
// ---------------------------------------------------------------
template <int K, int NC, int MODE>
__global__ __launch_bounds__(256) void gemm_wmma_f32(
    const float* __restrict__ A,
    const float* __restrict__ W,
    const float* __restrict__ bias,
    float* __restrict__ C,
    int M,
    const float* __restrict__ resid)
{
    constexpr int P = K + 4;                 // LDS row pitch (kills bank conflicts)
    __shared__ float At[16 * P];
    const int tid  = threadIdx.x;
    const int wave = tid >> 5;
    const int lane = tid & 31;
    const int tileM = blockIdx.x;

    // cooperative, coalesced load of the 16 x K A-tile (K is a power of 2)
#pragma unroll
    for (int idx = tid; idx < 16 * K; idx += 256) {
        int r = idx / K;
        int c = idx & (K - 1);
        int gm = tileM * 16 + r;
        At[r * P + c] = (gm < M) ? A[(size_t)gm * K + c] : 0.0f;
    }
    __syncthreads();

    const int nloc = lane & 15;       // B column / A row held by this lane
    const int hi   = lane >> 4;       // half-wave select
    const int kb   = hi << 1;         // K sub-offset (0 or 2)
    const int n    = blockIdx.y * 128 + wave * 16 + nloc;

    float bsc = bias[n];
    v8f acc;
#pragma unroll
    for (int r = 0; r < 8; ++r) acc[r] = bsc;

    // per-lane, loop-invariant bases; all inner-loop offsets are compile-time
    const float* __restrict__ wp0 = W + (size_t)kb * NC + n;
    const float* __restrict__ ap0 = &At[nloc * P + kb];

    for (int k0 = 0; k0 < K; k0 += 64) {
        const float* __restrict__ wp = wp0 + (size_t)k0 * NC;
        const float* __restrict__ ap = ap0 + k0;
#pragma unroll
        for (int ki = 0; ki < 64; ki += 4) {
            v2f a, b;
            a[0] = ap[ki];
            a[1] = ap[ki + 1];
            b[0] = wp[ki * NC];
            b[1] = wp[(ki + 1) * NC];
            acc = __builtin_amdgcn_wmma_f32_16x16x4_f32(
                false, a, false, b, (short)0, acc, false, false);
        }
    }

#pragma unroll
    for (int r = 0; r < 8; ++r) {
        int m = tileM * 16 + r + (hi << 3);
        if (m < M) {
            float val = acc[r];
            if (MODE == 1) {
                val += resid[(size_t)m * D_DIM + n];
                val = (val > 0.0f) ? val : expm1f(val);
            }
            C[(size_t)m * NC + n] = val;
        }
    }
}

// ---------------------------------------------------------------
// Pass A: per-edge/head attention logits + segment max (wave/edge)
// ---------------------------------------------------------------
__global__ __launch_bounds__(256) void edge_alpha_kernel(
    const int* __restrict__ ei, int E,
    const float* __restrict__ q, const float* __restrict__ k,
    const float* __restrict__ ebuf,
    float* __restrict__ alpha, unsigned* __restrict__ amaxU)
{
    const int wave = threadIdx.x >> 5;
    const int lane = threadIdx.x & 31;
    const int e = blockIdx.x * 8 + wave;
    if (e >= E) return;                        // wave-uniform exit
    const int src = ei[e];
    const int dst = ei[E + e];

    const float4* q4 = (const float4*)(q    + (size_t)dst * HC_DIM);
    const float4* k4 = (const float4*)(k    + (size_t)src * HC_DIM);
    const float4* e4 = (const float4*)(ebuf + (size_t)e   * HC_DIM);

    float partial = 0.0f;
    const int base = lane * 4;                 // lane covers dims [16*lane,16*lane+16)
#pragma unroll
    for (int j = 0; j < 4; ++j) {
        float4 qq = q4[base + j];
        float4 kk = k4[base + j];
        float4 ee = e4[base + j];
        partial += qq.x * (kk.x + ee.x) + qq.y * (kk.y + ee.y)
                 + qq.z * (kk.z + ee.z) + qq.w * (kk.w + ee.w);
    }
    // reduce across the 8 lanes of each head
    partial += __shfl_xor(partial, 1);
    partial += __shfl_xor(partial, 2);
    partial += __shfl_xor(partial, 4);

    if ((lane & 7) == 0) {
        int h = lane >> 3;
        float a = partial * 0.08838834764831845f;   // 1/sqrt(128)
        alpha[(size_t)e * H_HEADS + h] = a;
        atomicMax(&amaxU[dst * H_HEADS + h], encOrd(a));
    }
}

// ---------------------------------------------------------------
// Pass B: ex = exp(alpha - max); segment-sum denominator
// ---------------------------------------------------------------
__global__ __launch_bounds__(256) void edge_exp_kernel(
    const int* __restrict__ ei, int E,
    float* __restrict__ alpha, const unsigned* __restrict__ amaxU,
    float* __restrict__ denom)
{
    int t = blockIdx.x * blockDim.x + threadIdx.x;
    if (t >= E * H_HEADS) return;
    int e = t >> 2;
    int h = t & 3;
    int dst = ei[E + e];
    float m = decOrd(amaxU[dst * H_HEADS + h]);
    float ex = __expf(alpha[t] - m);
    alpha[t] = ex;
    atomicAdd(&denom[dst * H_HEADS + h], ex);
}

// ---------------------------------------------------------------
// Pass C: msg = (v[src]+e)*attn scatter-added to agg[dst]
// ---------------------------------------------------------------
__global__ __launch_bounds__(256) void edge_scatter_kernel(
    const int* __restrict__ ei, int E,
    const float* __restrict__ v, const float* __restrict__ ebuf,
    const float* __restrict__ exA, const float* __restrict__ denom,
    float* __restrict__ agg)
{
    const int e = blockIdx.x;
    const int src = ei[e];
    const int dst = ei[E + e];
    for (int i = threadIdx.x; i < HC_DIM; i += 256) {
        int h = i >> 7;
        float attn = exA[(size_t)e * H_HEADS + h] / denom[dst * H_HEADS + h];
        float val = (v[(size_t)src * HC_DIM + i] + ebuf[(size_t)e * HC_DIM + i]) * attn;
        atomicAdd(&agg[(size_t)dst * HC_DIM + i], val);
    }
}

// ---------------------------------------------------------------
// Skip-add + LayerNorm(512), one wave per node (wave32-native)
// ---------------------------------------------------------------
__global__ __launch_bounds__(256) void ln_kernel(
    const float* __restrict__ agg, const float* __restrict__ skip,
    const float* __restrict__ g, const float* __restrict__ b,
    float* __restrict__ normed, int N)
{
    const int wave = threadIdx.x >> 5;
    const int lane = threadIdx.x & 31;
    const int n = blockIdx.x * 8 + wave;
    if (n >= N) return;

    const float4* a4 = (const float4*)(agg  + (size_t)n * HC_DIM);
    const float4* s4 = (const float4*)(skip + (size_t)n * HC_DIM);
    float4 vbuf[4];
    float s = 0.0f, ss = 0.0f;
    const int base = lane * 4;
#pragma unroll
    for (int j = 0; j < 4; ++j) {
        float4 a = a4[base + j];
        float4 sk = s4[base + j];
        float4 t = make_float4(a.x + sk.x, a.y + sk.y, a.z + sk.z, a.w + sk.w);
        vbuf[j] = t;
        s  += t.x + t.y + t.z + t.w;
        ss += t.x * t.x + t.y * t.y + t.z * t.z + t.w * t.w;
    }
#pragma unroll
    for (int off = 16; off >= 1; off >>= 1) {
        s  += __shfl_xor(s, off);
        ss += __shfl_xor(ss, off);
    }
    float mean = s * (1.0f / 512.0f);
    float var  = ss * (1.0f / 512.0f) - mean * mean;
    float rstd = rsqrtf(var + 1e-5f);

    float4* o4 = (float4*)(normed + (size_t)n * HC_DIM);
    const float4* g4 = (const float4*)g;
    const float4* b4 = (const float4*)b;
#pragma unroll
    for (int j = 0; j < 4; ++j) {
        float4 t = vbuf[j];
        float4 gg = g4[base + j];
        float4 bb = b4[base + j];
        float4 o;
        o.x = (t.x - mean) * rstd * gg.x + bb.x;
        o.y = (t.y - mean) * rstd * gg.y + bb.y;
        o.z = (t.z - mean) * rstd * gg.z + bb.z;
        o.w = (t.w - mean) * rstd * gg.w + bb.w;
        o4[base + j] = o;
    }
}

extern "C" void kernel_launch(void* const* d_in, const int* in_sizes, int n_in,
                              void* d_out, int out_size, void* d_ws, size_t ws_size,
                              hipStream_t stream) {
    const float* x     = (const float*)d_in[0];
    const int*   ei    = (const int*)d_in[1];
    const float* ea    = (const float*)d_in[2];
    const float* Wq    = (const float*)d_in[3];
    const float* bq    = (const float*)d_in[4];
    const float* Wk    = (const float*)d_in[5];
    const float* bk    = (const float*)d_in[6];
    const float* Wv    = (const float*)d_in[7];
    const float* bv    = (const float*)d_in[8];
    const float* We    = (const float*)d_in[9];
    const float* be    = (const float*)d_in[10];
    const float* Wskip = (const float*)d_in[11];
    const float* bskip = (const float*)d_in[12];
    const float* ln_g  = (const float*)d_in[13];
    const float* ln_b  = (const float*)d_in[14];
    const float* Wlin  = (const float*)d_in[15];
    const float* blin  = (const float*)d_in[16];

    const int N = in_sizes[0] / D_DIM;   // 10000
    const int E = in_sizes[2] / D_DIM;   // 160000

    // ---- workspace carve (fp32) ----
    float* ws = (float*)d_ws;
    const size_t nHC = (size_t)N * HC_DIM;
    float* qB  = ws;
    float* kB  = qB  + nHC;
    float* vB  = kB  + nHC;
    float* skB = vB  + nHC;
    float* agg = skB + nHC;
    float* nor = agg + nHC;
    float* eB  = nor + nHC;                         // [E,512]
    float* alB = eB  + (size_t)E * HC_DIM;          // [E,4]  logits -> exp
    unsigned* amx = (unsigned*)(alB + (size_t)E * H_HEADS);   // [N,4]
    float*    den = (float*)(amx + (size_t)N * H_HEADS);      // [N,4]

    hipMemsetAsync(agg, 0, nHC * sizeof(float), stream);
    hipMemsetAsync(amx, 0, (size_t)N * H_HEADS * sizeof(unsigned), stream);
    hipMemsetAsync(den, 0, (size_t)N * H_HEADS * sizeof(float), stream);

    dim3 blk(256);
    dim3 gNode((N + 15) / 16, HC_DIM / 128);
    dim3 gEdge((E + 15) / 16, HC_DIM / 128);

    // node projections (q,k,v,skip) and edge projection e
    gemm_wmma_f32<D_DIM, HC_DIM, 0><<<gNode, blk, 0, stream>>>(x,  Wq,    bq,    qB,  N, nullptr);
    gemm_wmma_f32<D_DIM, HC_DIM, 0><<<gNode, blk, 0, stream>>>(x,  Wk,    bk,    kB,  N, nullptr);
    gemm_wmma_f32<D_DIM, HC_DIM, 0><<<gNode, blk, 0, stream>>>(x,  Wv,    bv,    vB,  N, nullptr);
    gemm_wmma_f32<D_DIM, HC_DIM, 0><<<gNode, blk, 0, stream>>>(x,  Wskip, bskip, skB, N, nullptr);
    gemm_wmma_f32<D_DIM, HC_DIM, 0><<<gEdge, blk, 0, stream>>>(ea, We,    be,    eB,  E, nullptr);

    // attention softmax over incoming edges (3 passes)
    edge_alpha_kernel<<<(E + 7) / 8, blk, 0, stream>>>(ei, E, qB, kB, eB, alB, amx);
    edge_exp_kernel<<<((size_t)E * H_HEADS + 255) / 256, blk, 0, stream>>>(ei, E, alB, amx, den);
    edge_scatter_kernel<<<E, blk, 0, stream>>>(ei, E, vB, eB, alB, den, agg);

    // skip + layernorm
    ln_kernel<<<(N + 7) / 8, blk, 0, stream>>>(agg, skB, ln_g, ln_b, nor, N);

    // final projection fused with residual + ELU
    gemm_wmma_f32<HC_DIM, D_DIM, 1><<<dim3((N + 15) / 16, 1), blk, 0, stream>>>(
        nor, Wlin, blin, (float*)d_out, N, x);
}